// Net_10402410791347
// MI455X (gfx1250) — compile-verified
//
#include <hip/hip_runtime.h>

typedef __attribute__((ext_vector_type(2))) float v2f;
typedef __attribute__((ext_vector_type(8))) float v8f;

namespace {
constexpr int kDim = 64;            // embedding dim
constexpr int kRowsPerWave = 16;    // one 16x16 WMMA tile of dots per wave
constexpr int kWavesPerBlock = 8;   // 256 threads
constexpr int kRowsPerBlock = kRowsPerWave * kWavesPerBlock;  // 128
}

__global__ __launch_bounds__(256) void dot_sigmoid_wmma(
    const int2* __restrict__ xs,   // (batch, 2) int32 index pairs
    const float* __restrict__ fe0, // (VOCAB, 64) f32
    const float* __restrict__ fe1, // (VOCAB, 64) f32
    float* __restrict__ out)       // (batch,) f32
{
  // Per-wave staging: 16 rows x 64 f32 per table. 8 KB/wave, 64 KB/block.
  __shared__ __align__(16) float ldsA[kWavesPerBlock][kRowsPerWave * kDim];
  __shared__ __align__(16) float ldsB[kWavesPerBlock][kRowsPerWave * kDim];

  const int lane = threadIdx.x & 31;
  const int wave = threadIdx.x >> 5;
  const int rowBase = blockIdx.x * kRowsPerBlock + wave * kRowsPerWave;

  float* lA = ldsA[wave];
  float* lB = ldsB[wave];

  const int half = lane >> 4;  // 0: lanes 0-15, 1: lanes 16-31
  const int sub  = lane & 15;

  // ---- Gather 16 rows of each table into LDS ----
  // Iteration j: 16 lanes per row, rows (2j+half). Each global_load_b128
  // instruction reads two contiguous 256B embedding rows (perfect bursts).
#pragma unroll
  for (int j = 0; j < kRowsPerWave / 2; ++j) {
    const int r = 2 * j + half;
    const int2 p = xs[rowBase + r];
    const float4 a = ((const float4*)(fe0 + (size_t)p.x * kDim))[sub];
    const float4 b = ((const float4*)(fe1 + (size_t)p.y * kDim))[sub];
    ((float4*)(lA + r * kDim))[sub] = a;
    ((float4*)(lB + r * kDim))[sub] = b;
  }

  __syncthreads();

  // ---- 16 x v_wmma_f32_16x16x4_f32: C(16x16) += A(16x4) x B(4x16) ----
  // A layout (32-bit, 16x4): lane m (0-15) holds A[m][2h+0..1] in 2 VGPRs,
  // h = lane>>4. B (4x16) mirrors it with N = lane&15. Both map to the same
  // LDS address pattern on the row-major staged tiles -> ds_load_b64 each.
  v8f acc = {};
  const int koff = 2 * half;
#pragma unroll
  for (int c = 0; c < kDim / 4; ++c) {
    const int k = 4 * c + koff;
    const v2f av = *(const v2f*)(lA + sub * kDim + k);
    const v2f bv = *(const v2f*)(lB + sub * kDim + k);
    acc = __builtin_amdgcn_wmma_f32_16x16x4_f32(
        /*neg_a=*/false, av, /*neg_b=*/false, bv,
        /*c_mod=*/(short)0, acc, /*reuse_a=*/false, /*reuse_b=*/false);
  }

  __syncthreads();

  // ---- Extract diagonal D[i][i] = dot(a_i, b_i) via LDS bounce ----
  // C/D layout: VGPR v, lanes 0-15 -> (M=v, N=lane); lanes 16-31 -> (M=v+8).
#pragma unroll
  for (int v = 0; v < 8; ++v) {
    lA[(v + 8 * half) * 16 + sub] = acc[v];
  }

  __syncthreads();

  if (lane < 16) {
    const float d = lA[lane * 16 + lane];
    // sigmoid(d) = 1 / (1 + exp2(-d * log2(e)))  -> v_exp_f32 + divide
    const float e = __builtin_amdgcn_exp2f(d * -1.44269504088896340736f);
    out[rowBase + lane] = 1.0f / (1.0f + e);
  }
}

extern "C" void kernel_launch(void* const* d_in, const int* in_sizes, int n_in,
                              void* d_out, int out_size, void* d_ws, size_t ws_size,
                              hipStream_t stream) {
  const int2*  xs  = (const int2*)d_in[0];
  const float* fe0 = (const float*)d_in[1];
  const float* fe1 = (const float*)d_in[2];
  float* out = (float*)d_out;

  const int batch  = in_sizes[0] / 2;          // xs is (batch, 2)
  const int blocks = batch / kRowsPerBlock;    // 1048576 / 128 = 8192
  dot_sigmoid_wmma<<<blocks, 256, 0, stream>>>(xs, fe0, fe1, out);
}